// SocialGNN_43465069035984
// MI455X (gfx1250) — compile-verified
//
#include <hip/hip_runtime.h>
#include <hip/hip_bf16.h>
#include <stdint.h>

typedef __attribute__((ext_vector_type(2))) float v2f;
typedef __attribute__((ext_vector_type(8))) float v8f;

#define F_IN  48
#define F_HID 64

// ---------------- degree / norm ----------------

__global__ void k_init_deg(float* deg, int n) {
    int i = blockIdx.x * blockDim.x + threadIdx.x;
    if (i < n) deg[i] = 1.0f;                    // self-loop contribution
}

__global__ void k_count_deg(const long long* __restrict__ ei, float* deg, int E) {
    int e = blockIdx.x * blockDim.x + threadIdx.x;
    if (e < E) atomicAdd(&deg[(int)ei[E + e]], 1.0f);   // dst = ei[1][e]
}

__global__ void k_rsqrt(float* deg, int n) {
    int i = blockIdx.x * blockDim.x + threadIdx.x;
    if (i < n) deg[i] = rsqrtf(deg[i]);          // deg >= 1 always
}

// ---------------- layer-1 GEMM: hx = x @ W1 (FP32 WMMA 16x16x4) ----------------
// One wave computes a 16-row x 64-col stripe: 4 N-tiles x 12 K-steps.

__global__ void k_gemm1_wmma(const float* __restrict__ x,
                             const float* __restrict__ W1,
                             float* __restrict__ hx, int mTiles) {
    int wave = (blockIdx.x * blockDim.x + threadIdx.x) >> 5;   // uniform per wave
    if (wave >= mTiles) return;                                 // whole-wave exit: EXEC stays all-1s
    int lane = threadIdx.x & 31;
    int half = lane >> 4;          // 0 -> lanes 0-15, 1 -> lanes 16-31
    int l15  = lane & 15;
    int m0   = wave * 16;

    // A (16x4 f32) layout: lanes 0-15 hold K={0,1}, lanes 16-31 hold K={2,3}
    v2f a[12];
    const float* xrow = x + (size_t)(m0 + l15) * F_IN + half * 2;
#pragma unroll
    for (int kt = 0; kt < 12; ++kt) {
        a[kt].x = xrow[kt * 4 + 0];
        a[kt].y = xrow[kt * 4 + 1];
    }

#pragma unroll
    for (int nt = 0; nt < 4; ++nt) {
        int col = nt * 16 + l15;
        v8f c = {};
#pragma unroll
        for (int kt = 0; kt < 12; ++kt) {
            // B (4x16 f32) layout: v0 = rows K0 (lanes0-15)/K2 (lanes16-31), v1 = K1/K3
            int kb = kt * 4 + half * 2;
            v2f b;
            b.x = W1[(kb + 0) * F_HID + col];
            b.y = W1[(kb + 1) * F_HID + col];
            c = __builtin_amdgcn_wmma_f32_16x16x4_f32(
                    /*neg_a=*/false, a[kt], /*neg_b=*/false, b,
                    /*c_mod=*/(short)0, c, /*reuse_a=*/false, /*reuse_b=*/false);
        }
        // C/D 16x16 f32: VGPR v -> row m0 + half*8 + v, col = nt*16 + l15
#pragma unroll
        for (int v = 0; v < 8; ++v) {
            hx[(size_t)(m0 + half * 8 + v) * F_HID + col] = c[v];
        }
    }
}

// Scalar tail (only launched if N % 16 != 0; N=100000 -> never)
__global__ void k_gemm1_tail(const float* __restrict__ x, const float* __restrict__ W1,
                             float* __restrict__ hx, int rowBase, int n) {
    int t = blockIdx.x * blockDim.x + threadIdx.x;
    int row = rowBase + t / F_HID;
    int col = t % F_HID;
    if (row >= n) return;
    float acc = 0.f;
    for (int k = 0; k < F_IN; ++k) acc += x[(size_t)row * F_IN + k] * W1[k * F_HID + col];
    hx[(size_t)row * F_HID + col] = acc;
}

// ---------------- layer-1 aggregate ----------------

__global__ void k_self1(const float* __restrict__ hx, const float* __restrict__ dinv,
                        float* __restrict__ agg, int n) {
    int idx = blockIdx.x * blockDim.x + threadIdx.x;
    if (idx >= n * F_HID) return;
    int i = idx >> 6;
    float d2 = dinv[i] * dinv[i];
    agg[idx] = hx[idx] * d2;                      // self-loop seeds accumulator (no atomics)
}

__global__ void k_scatter1(const long long* __restrict__ ei,
                           const float* __restrict__ hx,
                           const float* __restrict__ dinv,
                           float* __restrict__ agg, int E) {
    int t = blockIdx.x * blockDim.x + threadIdx.x;
    int e = t >> 5;                                // 32 lanes cover 64 feats as float2
    if (e >= E) return;
    int f2 = (t & 31) * 2;
    int s = (int)ei[e];
    int d = (int)ei[E + e];
    float norm = dinv[s] * dinv[d];
    float2 m = *(const float2*)(hx + (size_t)s * F_HID + f2);
    atomicAdd(&agg[(size_t)d * F_HID + f2 + 0], m.x * norm);
    atomicAdd(&agg[(size_t)d * F_HID + f2 + 1], m.y * norm);
}

__global__ void k_relu_bias(float* agg, const float* __restrict__ b1, int n) {
    int idx = blockIdx.x * blockDim.x + threadIdx.x;
    if (idx >= n * F_HID) return;
    float v = agg[idx] + b1[idx & 63];
    agg[idx] = v > 0.f ? v : 0.f;
}

// ---------------- layer-2: h2 = h @ W2  (64 -> 2, VALU dot) ----------------

__global__ void k_gemm2(const float* __restrict__ h, const float* __restrict__ W2,
                        float* __restrict__ h2, int n) {
    int i = blockIdx.x * blockDim.x + threadIdx.x;
    if (i >= n) return;
    const float4* hp = (const float4*)(h + (size_t)i * F_HID);
    float a0 = 0.f, a1 = 0.f;
#pragma unroll
    for (int q = 0; q < 16; ++q) {
        float4 hv = hp[q];
        int k = q * 4;
        a0 += hv.x * W2[(k + 0) * 2 + 0] + hv.y * W2[(k + 1) * 2 + 0]
            + hv.z * W2[(k + 2) * 2 + 0] + hv.w * W2[(k + 3) * 2 + 0];
        a1 += hv.x * W2[(k + 0) * 2 + 1] + hv.y * W2[(k + 1) * 2 + 1]
            + hv.z * W2[(k + 2) * 2 + 1] + hv.w * W2[(k + 3) * 2 + 1];
    }
    h2[(size_t)i * 2 + 0] = a0;
    h2[(size_t)i * 2 + 1] = a1;
}

__global__ void k_self2(const float* __restrict__ h2, const float* __restrict__ dinv,
                        const float* __restrict__ b2, float* __restrict__ out, int n) {
    int i = blockIdx.x * blockDim.x + threadIdx.x;
    if (i >= n) return;
    float d2 = dinv[i] * dinv[i];
    out[(size_t)i * 2 + 0] = h2[(size_t)i * 2 + 0] * d2 + b2[0];
    out[(size_t)i * 2 + 1] = h2[(size_t)i * 2 + 1] * d2 + b2[1];
}

__global__ void k_scatter2(const long long* __restrict__ ei,
                           const float* __restrict__ h2,
                           const float* __restrict__ dinv,
                           float* __restrict__ out, int E) {
    int e = blockIdx.x * blockDim.x + threadIdx.x;
    if (e >= E) return;
    int s = (int)ei[e];
    int d = (int)ei[E + e];
    float norm = dinv[s] * dinv[d];
    float2 m = *(const float2*)(h2 + (size_t)s * 2);
    atomicAdd(&out[(size_t)d * 2 + 0], m.x * norm);
    atomicAdd(&out[(size_t)d * 2 + 1], m.y * norm);
}

// ---------------- launch ----------------

extern "C" void kernel_launch(void* const* d_in, const int* in_sizes, int n_in,
                              void* d_out, int out_size, void* d_ws, size_t ws_size,
                              hipStream_t stream) {
    const float*     x   = (const float*)d_in[0];
    const long long* ei  = (const long long*)d_in[1];   // int64 edge_index [2,E]
    const float*     W1  = (const float*)d_in[2];
    const float*     b1  = (const float*)d_in[3];
    const float*     W2  = (const float*)d_in[4];
    const float*     b2  = (const float*)d_in[5];
    float*           out = (float*)d_out;

    const int N = in_sizes[0] / F_IN;
    const int E = in_sizes[1] / 2;

    // workspace layout: deg/dinv [N] | hx [N*64] | agg [N*64]; h2 reuses hx region
    float* deg = (float*)d_ws;
    float* hx  = deg + N;
    float* agg = hx + (size_t)N * F_HID;
    float* h2  = hx;   // hx dead after scatter1; reuse for [N*2]

    const int B = 256;
    auto cdiv = [](long long a, long long b) { return (unsigned)((a + b - 1) / b); };

    // degrees -> dinv
    k_init_deg <<<cdiv(N, B), B, 0, stream>>>(deg, N);
    k_count_deg<<<cdiv(E, B), B, 0, stream>>>(ei, deg, E);
    k_rsqrt    <<<cdiv(N, B), B, 0, stream>>>(deg, N);

    // layer 1 GEMM via FP32 WMMA (16x16x4), one wave per 16-row stripe
    int mTiles = N / 16;
    if (mTiles > 0)
        k_gemm1_wmma<<<cdiv((long long)mTiles * 32, B), B, 0, stream>>>(x, W1, hx, mTiles);
    int rem = N - mTiles * 16;
    if (rem > 0)
        k_gemm1_tail<<<cdiv((long long)rem * F_HID, B), B, 0, stream>>>(x, W1, hx, mTiles * 16, N);

    // layer 1 aggregate + activation
    k_self1    <<<cdiv((long long)N * F_HID, B), B, 0, stream>>>(hx, deg, agg, N);
    k_scatter1 <<<cdiv((long long)E * 32, B), B, 0, stream>>>(ei, hx, deg, agg, E);
    k_relu_bias<<<cdiv((long long)N * F_HID, B), B, 0, stream>>>(agg, b1, N);

    // layer 2
    k_gemm2    <<<cdiv(N, B), B, 0, stream>>>(agg, W2, h2, N);
    k_self2    <<<cdiv(N, B), B, 0, stream>>>(h2, deg, b2, out, N);
    k_scatter2 <<<cdiv(E, B), B, 0, stream>>>(ei, h2, deg, out, E);
}